// RasterizePointsXYsBlending_70557722739481
// MI455X (gfx1250) — compile-verified
//
#include <hip/hip_runtime.h>
#include <stdint.h>

namespace {
constexpr int kB = 8, kH = 256, kW = 256, kK = 8, kC = 64;
constexpr int kHW = kH * kW;                      // 65536
constexpr long long kNPix = (long long)kB * kHW;  // 524288
constexpr int kTPB = 256;                         // pixels per block (8 waves)
constexpr float kRNDC = (1.5f / 256.0f) * 2.0f;   // radius in NDC
constexpr float kInvR2 = 1.0f / (kRNDC * kRNDC);  // 1 / radius^rad_pow
}  // namespace

// One pixel per lane. Inputs for the block's 256 pixels are staged to LDS with
// CDNA5 async global->LDS b128 copies (ASYNCcnt). Feature rows are gathered
// from L2 (the whole 134MB table fits in the 192MB L2). Output is written with
// non-temporal stores so the streaming 134MB output does not evict features.
__global__ __launch_bounds__(kTPB) void raster_xys_blend_kernel(
    const float* __restrict__ dist,   // [B,H,W,K]
    const float* __restrict__ zbuf,   // [B,H,W,K]
    const int* __restrict__ pidx,     // [B,H,W,K]
    const float* __restrict__ feat,   // [N,C]
    float* __restrict__ out)          // [B,C,H,W]
{
  __shared__ float s_dist[kTPB * kK];  // 8 KB
  __shared__ float s_z[kTPB * kK];     // 8 KB
  __shared__ int s_idx[kTPB * kK];     // 8 KB

  const unsigned t = threadIdx.x;
  const size_t tileBase = (size_t)blockIdx.x * (kTPB * kK);

  // ---- async stage: 3 x 8KB tiles, 2 x b128 per lane per tile -------------
  {
    const char* gd = (const char*)(dist + tileBase);
    const char* gz = (const char*)(zbuf + tileBase);
    const char* gi = (const char*)(pidx + tileBase);
    const unsigned ld = (unsigned)(uintptr_t)&s_dist[0];
    const unsigned lz = (unsigned)(uintptr_t)&s_z[0];
    const unsigned li = (unsigned)(uintptr_t)&s_idx[0];
    const unsigned o0 = t * 16u;
    const unsigned o1 = (t + kTPB) * 16u;
    asm volatile("global_load_async_to_lds_b128 %0, %1, off" ::"v"(ld + o0),
                 "v"(gd + o0)
                 : "memory");
    asm volatile("global_load_async_to_lds_b128 %0, %1, off" ::"v"(ld + o1),
                 "v"(gd + o1)
                 : "memory");
    asm volatile("global_load_async_to_lds_b128 %0, %1, off" ::"v"(lz + o0),
                 "v"(gz + o0)
                 : "memory");
    asm volatile("global_load_async_to_lds_b128 %0, %1, off" ::"v"(lz + o1),
                 "v"(gz + o1)
                 : "memory");
    asm volatile("global_load_async_to_lds_b128 %0, %1, off" ::"v"(li + o0),
                 "v"(gi + o0)
                 : "memory");
    asm volatile("global_load_async_to_lds_b128 %0, %1, off" ::"v"(li + o1),
                 "v"(gi + o1)
                 : "memory");
    asm volatile("s_wait_asynccnt 0" ::: "memory");
  }
  __syncthreads();

  // ---- per-pixel slot data from LDS (b128 reads) --------------------------
  float dk[kK], zk[kK];
  int ik[kK];
  {
    const float4* sd = (const float4*)&s_dist[t * kK];
    const float4* sz = (const float4*)&s_z[t * kK];
    const int4* si = (const int4*)&s_idx[t * kK];
    *(float4*)&dk[0] = sd[0];
    *(float4*)&dk[4] = sd[1];
    *(float4*)&zk[0] = sz[0];
    *(float4*)&zk[4] = sz[1];
    *(int4*)&ik[0] = si[0];
    *(int4*)&ik[4] = si[1];
  }

  // ---- compositing weights: w_k = a_k * prod_{j<k}(1-a_j) -----------------
  float wk[kK];
  float trans = 1.0f;
#pragma unroll
  for (int k = 0; k < kK; ++k) {
    float d = dk[k] * kInvR2;
    d = fminf(fmaxf(d, 0.001f), 1.0f);
    float a = 1.0f - __builtin_sqrtf(d);
    const bool valid = (ik[k] >= 0) && (zk[k] >= 0.0f);
    a = valid ? a : 0.0f;
    wk[k] = a * trans;
    trans *= (1.0f - a);
    if (ik[k] < 0) ik[k] = 0;  // safe_idx (weight already zeroed)
  }

  const size_t p = (size_t)blockIdx.x * kTPB + t;  // global pixel id
  const int b = (int)(p / kHW);
  const int hw = (int)(p % kHW);
  float* obase = out + (size_t)b * ((size_t)kC * kHW) + hw;

#define FMA4(acc, f)                \
  do {                              \
    acc.x = fmaf(wv, (f).x, acc.x); \
    acc.y = fmaf(wv, (f).y, acc.y); \
    acc.z = fmaf(wv, (f).z, acc.z); \
    acc.w = fmaf(wv, (f).w, acc.w); \
  } while (0)

  // ---- gather + accumulate, 16 channels per chunk -------------------------
#pragma unroll
  for (int cc = 0; cc < kC / 16; ++cc) {
    float4 a0 = make_float4(0.f, 0.f, 0.f, 0.f), a1 = a0, a2 = a0, a3 = a0;
#pragma unroll
    for (int k = 0; k < kK; ++k) {
      const float wv = wk[k];
      if (wv != 0.0f) {  // skip dead slots: saves L2 gather bandwidth
        const float4* fr =
            (const float4*)(feat + (size_t)ik[k] * kC + cc * 16);
        const float4 f0 = fr[0], f1 = fr[1], f2 = fr[2], f3 = fr[3];
        FMA4(a0, f0);
        FMA4(a1, f1);
        FMA4(a2, f2);
        FMA4(a3, f3);
      }
    }
    // out[b, c, h, w]: per channel, 32 lanes write one contiguous 128B line.
    float* op = obase + (size_t)(cc * 16) * kHW;
    __builtin_nontemporal_store(a0.x, op + 0 * (size_t)kHW);
    __builtin_nontemporal_store(a0.y, op + 1 * (size_t)kHW);
    __builtin_nontemporal_store(a0.z, op + 2 * (size_t)kHW);
    __builtin_nontemporal_store(a0.w, op + 3 * (size_t)kHW);
    __builtin_nontemporal_store(a1.x, op + 4 * (size_t)kHW);
    __builtin_nontemporal_store(a1.y, op + 5 * (size_t)kHW);
    __builtin_nontemporal_store(a1.z, op + 6 * (size_t)kHW);
    __builtin_nontemporal_store(a1.w, op + 7 * (size_t)kHW);
    __builtin_nontemporal_store(a2.x, op + 8 * (size_t)kHW);
    __builtin_nontemporal_store(a2.y, op + 9 * (size_t)kHW);
    __builtin_nontemporal_store(a2.z, op + 10 * (size_t)kHW);
    __builtin_nontemporal_store(a2.w, op + 11 * (size_t)kHW);
    __builtin_nontemporal_store(a3.x, op + 12 * (size_t)kHW);
    __builtin_nontemporal_store(a3.y, op + 13 * (size_t)kHW);
    __builtin_nontemporal_store(a3.z, op + 14 * (size_t)kHW);
    __builtin_nontemporal_store(a3.w, op + 15 * (size_t)kHW);
  }
#undef FMA4
}

extern "C" void kernel_launch(void* const* d_in, const int* in_sizes, int n_in,
                              void* d_out, int out_size, void* d_ws,
                              size_t ws_size, hipStream_t stream) {
  (void)in_sizes;
  (void)n_in;
  (void)out_size;
  (void)d_ws;
  (void)ws_size;
  const float* dist = (const float*)d_in[0];
  const float* zbuf = (const float*)d_in[1];
  const int* pidx = (const int*)d_in[2];
  const float* feat = (const float*)d_in[3];
  float* out = (float*)d_out;

  const int nBlocks = (int)(kNPix / kTPB);  // 2048
  raster_xys_blend_kernel<<<nBlocks, kTPB, 0, stream>>>(dist, zbuf, pidx, feat,
                                                        out);
}